// ConstrainedDecoder_78254304133131
// MI455X (gfx1250) — compile-verified
//
#include <hip/hip_runtime.h>
#include <float.h>

// ---------------------------------------------------------------------------
// Constrained Viterbi decoder for MI455X (gfx1250).
//  - Forward max-plus recurrence with 6-deep async global->LDS tile pipeline
//    (gfx1250 GLOBAL_LOAD_ASYNC_TO_LDS_B32 + s_wait_asynccnt). 6 stages keep
//    <=40 of 63 ASYNCcnt outstanding while hiding ~5 steps of HBM latency.
//  - Async loads scatter the 45x45 f32 tile *transposed* into LDS (odd row
//    stride 49 -> bank-conflict-free column access).
//  - Backpointers (uint8) stream to workspace; backtrack runs in-kernel with
//    LDS-staged chunks to avoid dependent HBM pointer chasing.
// ---------------------------------------------------------------------------

#define GLOBAL_AS __attribute__((address_space(1)))
#define LDS_AS    __attribute__((address_space(3)))

#if defined(__has_builtin)
#  if __has_builtin(__builtin_amdgcn_global_load_async_to_lds_b32)
#    define ASYNC_LOAD_OK 1
#  endif
#  if __has_builtin(__builtin_amdgcn_s_wait_asynccnt)
#    define WAIT_ASYNC_OK 1
#  endif
#endif

namespace {

constexpr int   kB       = 16;
constexpr int   kT       = 1024;
constexpr int   kN       = 45;
constexpr int   kNN      = kN * kN;            // 2025
constexpr int   kStride  = 49;                 // odd -> conflict-free LDS columns
constexpr int   kTileF   = kN * kStride;       // 2205 floats per stage
constexpr int   kStages  = 6;                  // 52.9 KB of LDS tiles
constexpr int   kThreads = 256;
constexpr int   kLPT     = (kNN + kThreads - 1) / kThreads;  // 8 loads/thread/tile
constexpr float kNinf    = -1e5f;              // must match reference NINF

template <int N>
__device__ __forceinline__ void wait_async() {
#if defined(WAIT_ASYNC_OK)
  __builtin_amdgcn_s_wait_asynccnt(N);
#elif defined(ASYNC_LOAD_OK)
  asm volatile("s_wait_asynccnt %0" ::"n"(N) : "memory");
#endif
}

// Wait until tile t is resident: <= 8 async instrs per wave per in-flight tile.
__device__ __forceinline__ void wait_tile(int ahead) {
  if (ahead >= 5)      wait_async<40>();
  else if (ahead == 4) wait_async<32>();
  else if (ahead == 3) wait_async<24>();
  else if (ahead == 2) wait_async<16>();
  else if (ahead == 1) wait_async<8>();
  else                 wait_async<0>();
}

// Issue one 45x45 f32 tile: global (row-major i,j) -> LDS transposed [j][i].
__device__ __forceinline__ void issue_tile(const float* __restrict__ g,
                                           float* s,
                                           const int* __restrict__ eoff,
                                           const int* __restrict__ loff) {
#pragma unroll
  for (int k = 0; k < kLPT; ++k) {
    int e = eoff[k];
    if (e < kNN) {
#if defined(ASYNC_LOAD_OK)
      // builtin signature: (AS1 int* gsrc, AS3 int* ldst, imm offset, imm cpol)
      __builtin_amdgcn_global_load_async_to_lds_b32(
          (GLOBAL_AS int*)(g + e),
          (LDS_AS int*)(s + loff[k]), 0, 0);
#else
      s[loff[k]] = g[e];  // fallback: load + ds_store (compiler inserts waits)
#endif
    }
  }
}

}  // namespace

__global__ __launch_bounds__(kThreads)
void ConstrainedDecoder_78254304133131_kernel(
    const float* __restrict__ lp,          // [B,T,N,N]
    const int* __restrict__ lengths,       // [B]
    const unsigned char* __restrict__ startc,  // [N]  bool8
    const unsigned char* __restrict__ endc,    // [N]  bool8
    const unsigned char* __restrict__ transc,  // [N,N] bool8
    int* __restrict__ out,                 // [B,T]
    unsigned char* __restrict__ bp)        // workspace: [B,T,N] backpointers
{
  __shared__ float s_tile[kStages * kTileF];   // 52.9 KB, 6-stage pipeline
  __shared__ float s_alpha[48];                // padded to 48
  __shared__ float s_pv[4 * 48];               // per-ig partial max
  __shared__ int   s_pi[4 * 48];               // per-ig partial argmax
  __shared__ int   s_misc[2];                  // backtrack handoff (tag, t)

  const int b   = blockIdx.x;
  const int tid = threadIdx.x;
  const int ig  = tid >> 6;        // i-group 0..3 (i ranges 0-11,12-23,24-35,36-44)
  const int j   = tid & 63;        // output tag lane (active when j < 45)
  const int len = lengths[b];      // in [512, 1024]

  const float*   gmat = lp + (size_t)b * kT * kNN;
  unsigned char* bpb  = bp + (size_t)b * kT * kN;

  // Per-thread tile element offsets (t-invariant): element e -> LDS [j][i].
  int eoff[kLPT], loff[kLPT];
#pragma unroll
  for (int k = 0; k < kLPT; ++k) {
    int e = tid + k * kThreads;
    eoff[k] = e;
    int ii = e / kN, jj = e - ii * kN;
    loff[k] = jj * kStride + ii;
  }

  // LDS init: alpha (t=0 uses alpha=0), pad lanes poisoned low; tile row pads
  // zeroed once (async loads never touch columns 45..48).
  if (tid < 48) s_alpha[tid] = (tid < kN) ? 0.0f : -1e38f;
  for (int x = tid; x < kN * kStages; x += kThreads) {
    int jr = x % kN, st = x / kN;
    float* row = &s_tile[st * kTileF + jr * kStride];
    row[45] = 0.0f; row[46] = 0.0f; row[47] = 0.0f; row[48] = 0.0f;
  }

  // Constraint constants in registers.
  float tr[12];
  if (j < kN) {
#pragma unroll
    for (int k = 0; k < 12; ++k) {
      int i = ig * 12 + k;
      tr[k] = (i < kN) ? (transc[i * kN + j] ? 0.0f : kNinf) : 0.0f;
    }
  }
  float sadd = 0.0f, eadd = 0.0f;
  if (tid < kN) {
    sadd = startc[tid] ? 0.0f : kNinf;
    eadd = endc[tid]   ? 0.0f : kNinf;
  }

  // Prologue: fill the pipeline (len >= 512 > kStages always).
#pragma unroll
  for (int t = 0; t < kStages; ++t)
    issue_tile(gmat + (size_t)t * kNN, &s_tile[t * kTileF], eoff, loff);

  // ---------------- forward recurrence ----------------
  int stage = 0;
  for (int t = 0; t < len; ++t) {
    wait_tile(len - 1 - t);   // gate on oldest in-flight tile only
    __syncthreads();          // tile t resident for all waves; alpha published

    if (j < kN) {
      const float* row  = &s_tile[stage * kTileF + j * kStride];
      const float* ap   = &s_alpha[ig * 12];
      const float  tsel = (t > 0) ? 1.0f : 0.0f;  // no transition mask at t=0
      float acc = -3.0e38f;
      int   arg = 0;
#pragma unroll
      for (int k = 0; k < 12; ++k) {
        float cand = fmaf(tr[k], tsel, ap[k] + row[ig * 12 + k]);
        if (cand > acc) { acc = cand; arg = ig * 12 + k; }  // first-i wins ties
      }
      s_pv[ig * 48 + j] = acc;
      s_pi[ig * 48 + j] = arg;
    }
    __syncthreads();  // partials published; all tile-t reads finished

    if (tid < kN) {
      float acc = s_pv[tid];
      int   arg = s_pi[tid];
#pragma unroll
      for (int g = 1; g < 4; ++g) {
        float v = s_pv[g * 48 + tid];
        if (v > acc) { acc = v; arg = s_pi[g * 48 + tid]; }
      }
      if (t == 0)       acc += sadd;   // start mask on tag at position 0
      if (t == len - 1) acc += eadd;   // end mask on tag at last valid position
      s_alpha[tid] = acc;              // consumed after next top barrier
      if (t > 0) bpb[(size_t)t * kN + tid] = (unsigned char)arg;
    }

    // Refill this stage with tile t+kStages (WAR-safe: reads done pre-barrier2).
    if (t + kStages < len)
      issue_tile(gmat + (size_t)(t + kStages) * kNN,
                 &s_tile[stage * kTileF], eoff, loff);

    if (++stage == kStages) stage = 0;
  }

  __syncthreads();

  // Padding: positions >= len get PADDING_INDEX.
  for (int t = len + tid; t < kT; t += kThreads) out[(size_t)b * kT + t] = -1;

  // Make backpointer stores visible to this block's loads.
  __threadfence();
  __syncthreads();

  // ---------------- backtrack (LDS-staged, 2 chunks of 512 steps) -----------
  unsigned char* s_bp = (unsigned char*)s_tile;  // 23040 B <= 52920 B

  if (len > 512) {  // chunk A: t in [512, 1024)
    const unsigned int* src = (const unsigned int*)(bpb + 512 * kN);
    unsigned int*       dst = (unsigned int*)s_bp;
    for (int w = tid; w < (512 * kN) / 4; w += kThreads) dst[w] = src[w];
  }
  __syncthreads();

  if (tid == 0) {
    // argmax of final alpha (first index wins ties, like jnp.argmax)
    float best = s_alpha[0];
    int   jc   = 0;
    for (int q = 1; q < kN; ++q) {
      float v = s_alpha[q];
      if (v > best) { best = v; jc = q; }
    }
    int t = len - 1;
    while (t >= 512) {
      out[(size_t)b * kT + t] = jc;
      jc = s_bp[(t - 512) * kN + jc];
      --t;
    }
    s_misc[0] = jc;
    s_misc[1] = t;
  }
  __syncthreads();

  {  // chunk B: t in [0, 512)
    const unsigned int* src = (const unsigned int*)bpb;
    unsigned int*       dst = (unsigned int*)s_bp;
    for (int w = tid; w < (512 * kN) / 4; w += kThreads) dst[w] = src[w];
  }
  __syncthreads();

  if (tid == 0) {
    int jc = s_misc[0];
    int t  = s_misc[1];
    while (t >= 1) {
      out[(size_t)b * kT + t] = jc;
      jc = s_bp[t * kN + jc];
      --t;
    }
    out[(size_t)b * kT] = jc;  // tag at position 0
  }
}

extern "C" void kernel_launch(void* const* d_in, const int* in_sizes, int n_in,
                              void* d_out, int out_size, void* d_ws, size_t ws_size,
                              hipStream_t stream) {
  const float*         lp      = (const float*)d_in[0];          // [16,1024,45,45]
  const int*           lengths = (const int*)d_in[1];            // [16]
  const unsigned char* startc  = (const unsigned char*)d_in[2];  // [45] bool8
  const unsigned char* endc    = (const unsigned char*)d_in[3];  // [45] bool8
  const unsigned char* transc  = (const unsigned char*)d_in[4];  // [45,45] bool8
  int*                 out     = (int*)d_out;                    // [16,1024] int32
  unsigned char*       bp      = (unsigned char*)d_ws;           // 16*1024*45 = 720 KB

  (void)in_sizes; (void)n_in; (void)out_size; (void)ws_size;

  ConstrainedDecoder_78254304133131_kernel<<<dim3(kB), dim3(kThreads), 0, stream>>>(
      lp, lengths, startc, endc, transc, out, bp);
}